// FusedSliceWhereCatReplacement_60120952209915
// MI455X (gfx1250) — compile-verified
//
#include <hip/hip_runtime.h>
#include <stdint.h>

// ---------------------------------------------------------------------------
// FusedSliceWhereCat for MI455X (gfx1250)
//
// out (8192 x 8448 f32):
//   out[:, 0:256]    = unmatched
//   out[:, 256:8448] = where[row] ? slice[row, GATHER_IDX] : 0
// GATHER_IDX = 64 slices of width 128 starting every 256 columns
//   -> a 2D TDM tile: tile_dim0=128, tile_dim1=64, tensor_dim0_stride=256.
//
// Memory-bound: ~0.42 GB moved -> ~18 us at 23.3 TB/s. Strategy:
//   * TDM (tensor_load_to_lds) performs the strided gather into LDS (32 KB/row)
//   * nontemporal b128 stores stream the output (277 MB > 192 MB L2)
//   * where==0 rows skip the gather read entirely and store zeros
// ---------------------------------------------------------------------------

#define B_ROWS   8192
#define D_TOTAL  16384
#define D_UNM    256
#define WIDTH    128
#define STRIDE   256
#define SLICES   64
#define MATCHED  (SLICES * WIDTH)      // 8192
#define OUT_COLS (D_UNM + MATCHED)     // 8448

typedef uint32_t __attribute__((ext_vector_type(4))) u32x4;
typedef int      __attribute__((ext_vector_type(4))) i32x4;
typedef int      __attribute__((ext_vector_type(8))) i32x8;
typedef float    __attribute__((ext_vector_type(4))) f32x4;   // native vector for b128 ops

#if defined(__has_include)
#  if __has_include(<hip/amd_detail/amd_gfx1250_TDM.h>)
#    define TDM_SIX_ARG 1
#  endif
#endif

__global__ __launch_bounds__(256)
void FusedSliceWhereCat_kernel(const unsigned char* __restrict__ where_in,
                               const float* __restrict__ slice_in,
                               const float* __restrict__ unmatched,
                               float* __restrict__ out)
{
    const int row = blockIdx.x;
    const int tid = threadIdx.x;

    __shared__ float tile[MATCHED];    // 32 KB staging for the TDM gather

    float* out_row = out + (size_t)row * OUT_COLS;

    // ---- unmatched prefix: 256 floats = 64 float4s (threads 0..63) ----
    if (tid < D_UNM / 4) {
        const f32x4* src = (const f32x4*)(unmatched + (size_t)row * D_UNM);
        f32x4 v = __builtin_nontemporal_load(src + tid);
        __builtin_nontemporal_store(v, (f32x4*)out_row + tid);
    }

    const bool pred = (where_in[row] != 0);

    if (pred) {
        // ---- issue the TDM gather for this row (wave 0 only) ----
        if (tid == 0) {
            const uint64_t gaddr   = (uint64_t)(uintptr_t)(slice_in + (size_t)row * D_TOTAL);
            // Flat->LDS mapping truncates to addr[31:0] (ISA 10.2), so the low
            // 32 bits of the generic shared pointer are the LDS byte address.
            const uint32_t ldsaddr = (uint32_t)(uintptr_t)&tile[0];

            // D# group 0 (ISA 8.3): count=1 | lds_addr | global_addr | type=2
            u32x4 g0;
            g0.x = 1u;                                        // count=1, user desc
            g0.y = ldsaddr;                                   // lds_addr[31:0]
            g0.z = (uint32_t)(gaddr & 0xFFFFFFFFu);           // global_addr[31:0]
            g0.w = (uint32_t)((gaddr >> 32) & 0x01FFFFFFu)    // global_addr[56:32]
                 | (2u << 30);                                // type = 2 ("image")

            // D# group 1 (ISA 8.4): data_size=4B, dims/strides in elements
            i32x8 g1;
            g1[0] = (int)(2u << 16);                          // data_size=2 (4 bytes)
            g1[1] = (int)(((uint32_t)D_TOTAL & 0xFFFFu) << 16); // tensor_dim0[15:0]
            g1[2] = (int)(((uint32_t)D_TOTAL >> 16)           // tensor_dim0[31:16]
                 |        ((uint32_t)SLICES << 16));          // tensor_dim1[15:0]
            g1[3] = (int)((uint32_t)WIDTH << 16);             // tensor_dim1 hi | tile_dim0=128
            g1[4] = SLICES;                                   // tile_dim1=64, tile_dim2=0
            g1[5] = STRIDE;                                   // tensor_dim0_stride[31:0] = 256
            g1[6] = 0;                                        // stride hi | dim1_stride lo
            g1[7] = 0;

            i32x4 gz = {0, 0, 0, 0};                          // groups 2/3 unused (2D)
#ifdef TDM_SIX_ARG
            i32x8 gz8 = {0, 0, 0, 0, 0, 0, 0, 0};
            __builtin_amdgcn_tensor_load_to_lds(g0, g1, gz, gz, gz8, 0);
#else
            __builtin_amdgcn_tensor_load_to_lds(g0, g1, gz, gz, 0);
#endif
        }
        __builtin_amdgcn_s_wait_tensorcnt(0);   // wave 0 drains its DMA; no-op for others
        __syncthreads();                        // LDS tile visible to all 8 waves

        // ---- stream LDS tile -> out[row, 256:8448] (2048 float4s, 8/thread) ----
        f32x4*       dst = (f32x4*)(out_row + D_UNM);
        const f32x4* src = (const f32x4*)tile;
#pragma unroll
        for (int i = 0; i < MATCHED / 4 / 256; ++i) {
            f32x4 v = src[tid + i * 256];
            __builtin_nontemporal_store(v, dst + tid + i * 256);
        }
    } else {
        // ---- masked row: no gather read, just stream zeros ----
        const f32x4 z = {0.f, 0.f, 0.f, 0.f};
        f32x4* dst = (f32x4*)(out_row + D_UNM);
#pragma unroll
        for (int i = 0; i < MATCHED / 4 / 256; ++i) {
            __builtin_nontemporal_store(z, dst + tid + i * 256);
        }
    }
}

extern "C" void kernel_launch(void* const* d_in, const int* in_sizes, int n_in,
                              void* d_out, int out_size, void* d_ws, size_t ws_size,
                              hipStream_t stream)
{
    // setup_inputs order: where_input (bool, 8192), slice_input (f32, 8192*16384),
    // unmatched (f32, 8192*256), cat_dim (scalar), slice_dim (scalar)
    const unsigned char* where_in  = (const unsigned char*)d_in[0];
    const float*         slice_in  = (const float*)d_in[1];
    const float*         unmatched = (const float*)d_in[2];
    float*               out       = (float*)d_out;

    (void)in_sizes; (void)n_in; (void)out_size; (void)d_ws; (void)ws_size;

    dim3 grid(B_ROWS);
    dim3 block(256);
    FusedSliceWhereCat_kernel<<<grid, block, 0, stream>>>(where_in, slice_in, unmatched, out);
}